// RelationRNN_48610439856737
// MI455X (gfx1250) — compile-verified
//
#include <hip/hip_runtime.h>
#include <math.h>

typedef float v2f __attribute__((ext_vector_type(2)));
typedef float v8f __attribute__((ext_vector_type(8)));

#define N_EDGES 500000
#define N_RELS  256
#define DIM     256
#define N3      768   // 3*DIM

// ---------------------------------------------------------------------------
// Kernel 1: zero the accumulator region of the workspace (sums + counts).
// ---------------------------------------------------------------------------
__global__ void zero_kernel(float* __restrict__ p, int n) {
    int i = blockIdx.x * blockDim.x + threadIdx.x;
    if (i < n) p[i] = 0.0f;
}

// ---------------------------------------------------------------------------
// Kernel 2: LDS-staged segment sum.
// Each block holds the full 256x256 f32 bin table (256 KB) in dynamic LDS
// (fits in CDNA5's 320 KB WGP LDS). One wave processes one edge at a time:
// lane l covers channels [8l, 8l+8) via two float4 loads, then ds_add_f32
// into the per-block table. Table is flushed once with global atomics.
// ---------------------------------------------------------------------------
__global__ __launch_bounds__(256) void accum_kernel(
        const int* __restrict__ nid, const int* __restrict__ rel,
        const float* __restrict__ emb, float* __restrict__ gsum,
        float* __restrict__ counts, int do_count) {
    extern __shared__ float lsum[];                 // N_RELS*DIM floats
    for (int i = threadIdx.x; i < N_RELS * DIM; i += blockDim.x)
        lsum[i] = 0.0f;
    __syncthreads();

    const int lane  = threadIdx.x & 31;
    const int wave  = threadIdx.x >> 5;
    const int wpb   = blockDim.x >> 5;
    const int gw    = blockIdx.x * wpb + wave;      // global wave id
    const int nw    = gridDim.x * wpb;              // total waves

    for (int e = gw; e < N_EDGES; e += nw) {
        const int n = nid[e];
        const int r = rel[e];
        // prefetch the row this wave will touch next iteration
        if (e + nw < N_EDGES) {
            const int n2 = nid[e + nw];
            __builtin_prefetch(emb + (size_t)n2 * DIM + lane * 8, 0, 0);
        }
        const float4* row = (const float4*)(emb + (size_t)n * DIM);
        float4 v0 = row[lane * 2 + 0];
        float4 v1 = row[lane * 2 + 1];
        float* d = lsum + r * DIM + lane * 8;
        atomicAdd(d + 0, v0.x); atomicAdd(d + 1, v0.y);
        atomicAdd(d + 2, v0.z); atomicAdd(d + 3, v0.w);
        atomicAdd(d + 4, v1.x); atomicAdd(d + 5, v1.y);
        atomicAdd(d + 6, v1.z); atomicAdd(d + 7, v1.w);
        if (do_count && lane == 0) atomicAdd(counts + r, 1.0f);
    }
    __syncthreads();
    for (int i = threadIdx.x; i < N_RELS * DIM; i += blockDim.x)
        atomicAdd(gsum + i, lsum[i]);
}

// ---------------------------------------------------------------------------
// Kernel 3: in-place mean over both sum tables (src at 0, dst at 65536).
// ---------------------------------------------------------------------------
__global__ void mean_kernel(float* __restrict__ sums,
                            const float* __restrict__ counts, int n) {
    int idx = blockIdx.x * blockDim.x + threadIdx.x;
    if (idx >= n) return;
    int r = (idx >> 8) & (N_RELS - 1);
    sums[idx] = sums[idx] / fmaxf(counts[r], 1.0f);
}

// ---------------------------------------------------------------------------
// Kernel 4: C[256 x 768] = A[256 x 256] * W[768 x 256]^T + bias, f32 WMMA.
// NT GEMM via V_WMMA_F32_16X16X4_F32. One wave = one 16x16 C tile, K looped
// in steps of 4. A element stride is parameterized so the GRU hidden state
// can be read directly from dynamic_relation_emb (row stride 512, elem
// stride 2) without repacking.
//
// f32 16x16x4 A layout: lane%16 = M row; VGPR0/VGPR1 hold K = 2*(lane/16)+{0,1}.
// B (4x16) mirrored: lane%16 = N col; same K split. C tile: VGPR v ->
// row = 16*tm + v + 8*(lane/16), col = 16*tn + lane%16.
// Grid covers exactly 16*48 = 768 tiles -> EXEC all-1s (WMMA requirement).
// ---------------------------------------------------------------------------
__global__ __launch_bounds__(256) void gemm_wmma(
        const float* __restrict__ A, int a_rs, int a_es,
        const float* __restrict__ W, const float* __restrict__ bias,
        float* __restrict__ C) {
    const int lane = threadIdx.x & 31;
    const int wave = threadIdx.x >> 5;
    const int tile = blockIdx.x * 8 + wave;   // 96 blocks * 8 waves = 768
    const int tm   = tile / 48;
    const int tn   = tile % 48;
    const int half = lane >> 4;
    const int ml   = lane & 15;
    const int m    = tm * 16 + ml;
    const int n    = tn * 16 + ml;

    const float* arow = A + (size_t)m * a_rs;
    const float* wrow = W + (size_t)n * DIM;

    v8f acc = {};
    for (int k0 = 0; k0 < DIM; k0 += 4) {
        const int k = k0 + 2 * half;
        v2f a, b;
        a.x = arow[(size_t)k * a_es];
        a.y = arow[(size_t)(k + 1) * a_es];
        b.x = wrow[k];
        b.y = wrow[k + 1];
        acc = __builtin_amdgcn_wmma_f32_16x16x4_f32(
                /*neg_a=*/false, a, /*neg_b=*/false, b,
                /*c_mod=*/(short)0, acc, /*reuse_a=*/false, /*reuse_b=*/false);
    }
    const float bv = bias[n];
#pragma unroll
    for (int v = 0; v < 8; ++v) {
        const int row = tm * 16 + v + 8 * half;
        C[(size_t)row * N3 + n] = acc[v] + bv;
    }
}

// ---------------------------------------------------------------------------
// Kernel 5: GRU gates + interleaved (stride 2) output store.
// ---------------------------------------------------------------------------
__global__ void gru_gates(const float* __restrict__ gi,
                          const float* __restrict__ gh,
                          const float* __restrict__ dyn, int which,
                          float* __restrict__ out) {
    int idx = blockIdx.x * blockDim.x + threadIdx.x;   // 65536
    if (idx >= N_RELS * DIM) return;
    int r = idx >> 8;
    int d = idx & (DIM - 1);
    const float* gir = gi + (size_t)r * N3;
    const float* ghr = gh + (size_t)r * N3;
    float i_r = gir[d], i_z = gir[DIM + d], i_n = gir[2 * DIM + d];
    float h_r = ghr[d], h_z = ghr[DIM + d], h_n = ghr[2 * DIM + d];
    float h   = dyn[(size_t)r * 512 + d * 2 + which];
    float rg  = 1.0f / (1.0f + expf(-(i_r + h_r)));
    float z   = 1.0f / (1.0f + expf(-(i_z + h_z)));
    float ng  = tanhf(i_n + rg * h_n);
    out[(size_t)idx * 2 + which] = (1.0f - z) * ng + z * h;
}

// ---------------------------------------------------------------------------
extern "C" void kernel_launch(void* const* d_in, const int* in_sizes, int n_in,
                              void* d_out, int out_size, void* d_ws, size_t ws_size,
                              hipStream_t stream) {
    (void)in_sizes; (void)n_in; (void)out_size; (void)ws_size;

    const int*   src_nid = (const int*)d_in[0];
    const int*   dst_nid = (const int*)d_in[1];
    const int*   rel     = (const int*)d_in[2];
    const float* ent     = (const float*)d_in[3];
    const float* dyn     = (const float*)d_in[4];
    const float* W_ih    = (const float*)d_in[5];
    const float* W_hh    = (const float*)d_in[6];
    const float* b_ih    = (const float*)d_in[7];
    const float* b_hh    = (const float*)d_in[8];
    float*       out     = (float*)d_out;

    // workspace layout (floats)
    float* ws       = (float*)d_ws;
    float* sums_src = ws;                       // 65536  (becomes X_src)
    float* sums_dst = ws + 65536;               // 65536  (becomes X_dst)
    float* counts   = ws + 131072;              // 256
    float* gi_src   = ws + 131328;              // 196608
    float* gh_src   = gi_src + 196608;          // 196608
    float* gi_dst   = gh_src + 196608;          // 196608
    float* gh_dst   = gi_dst + 196608;          // 196608

    const int NACC = 131328;                    // sums + counts
    zero_kernel<<<(NACC + 255) / 256, 256, 0, stream>>>(ws, NACC);

    const size_t ldsBytes = (size_t)N_RELS * DIM * sizeof(float);   // 256 KB
    accum_kernel<<<64, 256, ldsBytes, stream>>>(src_nid, rel, ent,
                                                sums_src, counts, 1);
    accum_kernel<<<64, 256, ldsBytes, stream>>>(dst_nid, rel, ent,
                                                sums_dst, counts, 0);

    mean_kernel<<<(2 * 65536) / 256, 256, 0, stream>>>(ws, counts, 2 * 65536);

    // gi = X @ W_ih^T + b_ih ; gh = H @ W_hh^T + b_hh   (H strided in dyn)
    gemm_wmma<<<96, 256, 0, stream>>>(sums_src, DIM, 1, W_ih, b_ih, gi_src);
    gemm_wmma<<<96, 256, 0, stream>>>(dyn + 0,  512, 2, W_hh, b_hh, gh_src);
    gemm_wmma<<<96, 256, 0, stream>>>(sums_dst, DIM, 1, W_ih, b_ih, gi_dst);
    gemm_wmma<<<96, 256, 0, stream>>>(dyn + 1,  512, 2, W_hh, b_hh, gh_dst);

    gru_gates<<<65536 / 256, 256, 0, stream>>>(gi_src, gh_src, dyn, 0, out);
    gru_gates<<<65536 / 256, 256, 0, stream>>>(gi_dst, gh_dst, dyn, 1, out);
}